// NNModel3_4526895530076
// MI455X (gfx1250) — compile-verified
//
#include <hip/hip_runtime.h>
#include <stdint.h>

typedef __attribute__((ext_vector_type(2))) float v2f;
typedef __attribute__((ext_vector_type(8))) float v8f;

#define THREADS 256
#define TILE_E  64   // elements per block tile (64 * 256B = 16KB stiffness)

// ---------------- kernel 1: zero F and u1 ----------------
__global__ void fem_zero(float* __restrict__ u1, float* __restrict__ F, int ndof) {
    int i = blockIdx.x * blockDim.x + threadIdx.x;
    if (i < ndof) { u1[i] = 0.0f; F[i] = 0.0f; }
}

// ---------------- kernel 2: u1[bc[k]] = w1[k] * u[bc[k]] ----------------
__global__ void fem_bc(const float* __restrict__ u, const float* __restrict__ w1,
                       const int* __restrict__ bc, float* __restrict__ u1, int n) {
    int k = blockIdx.x * blockDim.x + threadIdx.x;
    if (k < n) {
        int idx = bc[k];
        u1[idx] = w1[k] * u[idx];
    }
}

// ---------------- kernel 3: gather -> per-element 8x8 matvec (WMMA) -> scatter-add ----------------
// Memory-bound at 0.4 flop/byte: stream 128MB of stiffness via async-to-LDS,
// do the einsum on v_wmma_f32_16x16x4_f32 (2 elements per WMMA: M rows 0..7 = e0,
// 8..15 = e1; B columns n<8 carry ue0, n>=8 carry ue1; diagonal blocks of D are
// the matvec results, replicated across the 8 columns of each half).
__global__ __launch_bounds__(THREADS) void fem_assemble(
        const float* __restrict__ u1,
        const int*   __restrict__ edof,   // [nelem][8] int32
        const float* __restrict__ Kmat,   // [nelem][8][8] f32, row-major
        float* __restrict__ F,
        int nelem)
{
    __shared__ alignas(16) float sK[TILE_E][8][8];   // 16 KB stiffness tile
    __shared__ alignas(32) float sU[TILE_E][8];      // gathered ue
    __shared__ alignas(32) float sFe[TILE_E][8];     // WMMA results bounce buffer
    __shared__ int sIdx[TILE_E][8];                  // scatter indices

    const int tid      = threadIdx.x;
    const int tileBase = blockIdx.x * TILE_E;

    // --- stage stiffness tile with async-to-LDS (ASYNCcnt path) ---
    {
        uint32_t ldsBase = (uint32_t)(size_t)(&sK[0][0][0]); // low 32 bits of generic ptr = LDS offset
        uint64_t gbase   = (uint64_t)(size_t)Kmat + (uint64_t)tileBase * 256ull;
        uint64_t limit   = (uint64_t)(size_t)Kmat + (uint64_t)nelem * 256ull - 16ull;
#pragma unroll
        for (int i = 0; i < 4; ++i) {
            uint32_t off = (uint32_t)tid * 16u + (uint32_t)i * 4096u;
            uint64_t ga  = gbase + (uint64_t)off;
            if (ga > limit) ga = limit;                  // clamp OOB tail lanes to a safe addr
            uint32_t la = ldsBase + off;
            asm volatile("global_load_async_to_lds_b128 %0, %1, off"
                         :: "v"(la), "v"(ga) : "memory");
        }
    }

    // --- stage gathered ue values and scatter indices (overlaps async loads) ---
#pragma unroll
    for (int s0 = 0; s0 < 2; ++s0) {
        int s  = tid + s0 * THREADS;          // 512 slots = TILE_E * 8
        int e  = s >> 3;
        int j  = s & 7;
        int ge = tileBase + e;
        int gi = 0; float uv = 0.0f;
        if (ge < nelem) {
            gi = edof[ge * 8 + j];
            uv = u1[gi];
        }
        sIdx[e][j] = gi;
        sU[e][j]   = uv;
    }

    asm volatile("s_wait_asynccnt 0x0" ::: "memory");   // this wave's async LDS writes done
    __syncthreads();                                    // all waves' writes visible

    const int lane = tid & 31;
    const int wave = tid >> 5;
    const int n16  = lane & 15;        // M (for A) / N (for B) index within 16
    const int half = lane >> 4;        // 0: lanes 0-15 (K=0,1), 1: lanes 16-31 (K=2,3)
    const int kk   = half * 2;
    const int mRow = n16 & 7;          // row within element
    // element slot for this lane's A-row and B-column (same mapping by construction)
    // computed per pair below.

#pragma unroll
    for (int p = 0; p < 4; ++p) {
        const int e0l = wave * 8 + p * 2;       // local pair base (uniform per wave)
        const int e0g = tileBase + e0l;
        if (e0g + 1 < nelem) {
            const int mElem = e0l + (n16 >> 3); // e0 for m/n<8, e1 otherwise
            v8f acc = {};
#pragma unroll
            for (int kc = 0; kc < 2; ++kc) {    // k-columns 0..3 then 4..7, accumulate via C
                const int koff = kc * 4;
                // A[m][k]: rows of K_e; lane half selects K pair {0,1} or {2,3} -> ds_load_b64
                v2f a = *(const v2f*)&sK[mElem][mRow][koff + kk];
                // B[k][n]: ue of elem(n) at k; VGPR0=K0/K2, VGPR1=K1/K3 -> ds_load_b64
                v2f b = *(const v2f*)&sU[mElem][koff + kk];
                acc = __builtin_amdgcn_wmma_f32_16x16x4_f32(
                        /*neg_a=*/false, a, /*neg_b=*/false, b,
                        /*c_mod=*/(short)0, acc, /*reuse_a=*/false, /*reuse_b=*/false);
            }
            // D layout: VGPR r, lanes0-15 -> (M=r, N=lane); lanes16-31 -> (M=r+8, N=lane-16)
            // lane 0  holds fe(e0)[r] in acc[r];  lane 24 holds fe(e1)[r] in acc[r].
            if (lane == 0)  *(v8f*)&sFe[e0l][0]     = acc;
            if (lane == 24) *(v8f*)&sFe[e0l + 1][0] = acc;
            // same-wave LDS ops are in-order: safe to read back immediately
            if (lane < 16) {
                const int el = e0l + (lane >> 3);
                const int r  = lane & 7;
                atomicAdd(&F[sIdx[el][r]], sFe[el][r]);
            }
        } else if (e0g < nelem) {
            // odd tail element: plain VALU matvec (8 lanes, one row each)
            if (lane < 8) {
                float fe = 0.0f;
#pragma unroll
                for (int j = 0; j < 8; ++j) fe += sK[e0l][lane][j] * sU[e0l][j];
                atomicAdd(&F[sIdx[e0l][lane]], fe);
            }
        }
    }
}

extern "C" void kernel_launch(void* const* d_in, const int* in_sizes, int n_in,
                              void* d_out, int out_size, void* d_ws, size_t ws_size,
                              hipStream_t stream) {
    const float* u    = (const float*)d_in[0];
    const float* w1   = (const float*)d_in[1];
    const int*   bc   = (const int*)d_in[2];
    const int*   edof = (const int*)d_in[3];
    const float* K    = (const float*)d_in[4];
    float* F  = (float*)d_out;
    float* u1 = (float*)d_ws;   // NDOF floats of scratch

    const int ndof  = in_sizes[0];
    const int nbc   = in_sizes[2];
    const int nelem = in_sizes[3] / 8;

    {
        int blocks = (ndof + THREADS - 1) / THREADS;
        fem_zero<<<blocks, THREADS, 0, stream>>>(u1, F, ndof);
    }
    {
        int blocks = (nbc + THREADS - 1) / THREADS;
        fem_bc<<<blocks, THREADS, 0, stream>>>(u, w1, bc, u1, nbc);
    }
    {
        int blocks = (nelem + TILE_E - 1) / TILE_E;
        fem_assemble<<<blocks, THREADS, 0, stream>>>(u1, edof, K, F, nelem);
    }
}